// RGCN_38560216384099
// MI455X (gfx1250) — compile-verified
//
#include <hip/hip_runtime.h>

#define N_NODES 100000
#define N_EDGES 1000000
#define D 64
#define LDP 68                 // padded LDS stride (floats) for transposed weights
#define NT (N_NODES / 16)      // 6250 row tiles

typedef float v2f __attribute__((ext_vector_type(2)));
typedef float v8f __attribute__((ext_vector_type(8)));

// ---------------------------------------------------------------- zero ws
__global__ void rgcn_zero(float* __restrict__ p, int n4) {
    int i = blockIdx.x * blockDim.x + threadIdx.x;
    if (i < n4) ((float4*)p)[i] = make_float4(0.f, 0.f, 0.f, 0.f);
}

// ----------------------------------------------- RGCN masked edge scatter
// one wave per edge; lane owns features [2*lane, 2*lane+1]
__global__ void rgcn_scatter(const float* __restrict__ x,
                             const int* __restrict__ ei,
                             const int* __restrict__ et,
                             float* __restrict__ sm,    // [2][N][64]
                             float* __restrict__ cnt) { // [2][N]
    int lane = threadIdx.x & 31;
    int e = (blockIdx.x * blockDim.x + threadIdx.x) >> 5;
    if (e >= N_EDGES) return;
    int src = ei[e];
    int dst = ei[N_EDGES + e];
    int r   = et[e];
    v2f v = *(const v2f*)(x + (size_t)src * D + lane * 2);
    float* sd = sm + ((size_t)r * N_NODES + dst) * D + lane * 2;
    atomicAdd(sd + 0, v.x);
    atomicAdd(sd + 1, v.y);
    if (lane == 0) atomicAdd(cnt + (size_t)r * N_NODES + dst, 1.0f);
}

// ------------------------------------------------ GraphConv edge scatter
__global__ void gc_scatter(const float* __restrict__ h,
                           const int* __restrict__ ei,
                           float* __restrict__ agg) {
    int lane = threadIdx.x & 31;
    int e = (blockIdx.x * blockDim.x + threadIdx.x) >> 5;
    if (e >= N_EDGES) return;
    int src = ei[e];
    int dst = ei[N_EDGES + e];
    v2f v = *(const v2f*)(h + (size_t)src * D + lane * 2);
    float* ad = agg + (size_t)dst * D + lane * 2;
    atomicAdd(ad + 0, v.x);
    atomicAdd(ad + 1, v.y);
}

// ------------------------------------------------------- RGCN WMMA GEMM
// h = relu(x@root + bias + (sm0/max(c0,1))@W0 + (sm1/max(c1,1))@W1)
__global__ __launch_bounds__(256) void rgcn_gemm(
    const float* __restrict__ x,
    const float* __restrict__ sm, const float* __restrict__ cnt,
    const float* __restrict__ Wrel,   // rgcn_weight [2][64][64]
    const float* __restrict__ Wroot,  // [64][64]
    const float* __restrict__ bias,   // [64]
    float* h) {
    __shared__ float lw[3 * D * LDP];   // transposed: lw[m][col*LDP + k]
    for (int i = threadIdx.x; i < 3 * D * D; i += 256) {
        int m = i / (D * D);
        int rem = i - m * D * D;
        int k = rem >> 6, n = rem & 63;
        const float* W = (m == 0) ? Wroot : (Wrel + (m - 1) * D * D);
        lw[m * D * LDP + n * LDP + k] = W[rem];
    }
    __syncthreads();

    int lane = threadIdx.x & 31;
    int tile = blockIdx.x * 8 + (threadIdx.x >> 5);
    if (tile >= NT) return;

    int row   = tile * 16 + (lane & 15);
    int klane = (lane >> 4) << 1;          // 0 or 2
    float inv0 = 1.0f / fmaxf(cnt[row], 1.0f);
    float inv1 = 1.0f / fmaxf(cnt[N_NODES + row], 1.0f);

    v8f acc[4];
#pragma unroll
    for (int t = 0; t < 4; ++t) {
        float bv = bias[t * 16 + (lane & 15)];
        acc[t] = (v8f){bv, bv, bv, bv, bv, bv, bv, bv};
    }

    const float* xrow  = x  + (size_t)row * D;
    const float* s0row = sm + (size_t)row * D;
    const float* s1row = sm + (size_t)N_NODES * D + (size_t)row * D;

    for (int k0 = 0; k0 < D; k0 += 4) {
        int kk = k0 + klane;
        v2f ax = *(const v2f*)(xrow + kk);
        v2f a0 = *(const v2f*)(s0row + kk); a0 = a0 * inv0;
        v2f a1 = *(const v2f*)(s1row + kk); a1 = a1 * inv1;
#pragma unroll
        for (int t = 0; t < 4; ++t) {
            int col = t * 16 + (lane & 15);
            v2f br = *(const v2f*)&lw[0 * D * LDP + col * LDP + kk];
            v2f b0 = *(const v2f*)&lw[1 * D * LDP + col * LDP + kk];
            v2f b1 = *(const v2f*)&lw[2 * D * LDP + col * LDP + kk];
            acc[t] = __builtin_amdgcn_wmma_f32_16x16x4_f32(false, ax, false, br, (short)0, acc[t], false, false);
            acc[t] = __builtin_amdgcn_wmma_f32_16x16x4_f32(false, a0, false, b0, (short)0, acc[t], false, false);
            acc[t] = __builtin_amdgcn_wmma_f32_16x16x4_f32(false, a1, false, b1, (short)0, acc[t], false, false);
        }
    }

    int mo = (lane >> 4) * 8;
#pragma unroll
    for (int t = 0; t < 4; ++t) {
        int col = t * 16 + (lane & 15);
#pragma unroll
        for (int v = 0; v < 8; ++v)
            h[(size_t)(tile * 16 + mo + v) * D + col] = fmaxf(acc[t][v], 0.0f);
    }
}

// -------------------------------------------------- GraphConv WMMA GEMM
// out = agg@Wrel + b_rel + h@Wroot   (out may alias h: waves only touch own tile)
__global__ __launch_bounds__(256) void gc_gemm(
    const float* __restrict__ agg, const float* h,
    const float* __restrict__ Wrel, const float* __restrict__ Wroot,
    const float* __restrict__ brel, float* out) {
    __shared__ float lw[2 * D * LDP];
    for (int i = threadIdx.x; i < 2 * D * D; i += 256) {
        int m = i / (D * D);
        int rem = i - m * D * D;
        int k = rem >> 6, n = rem & 63;
        const float* W = (m == 0) ? Wrel : Wroot;
        lw[m * D * LDP + n * LDP + k] = W[rem];
    }
    __syncthreads();

    int lane = threadIdx.x & 31;
    int tile = blockIdx.x * 8 + (threadIdx.x >> 5);
    if (tile >= NT) return;

    int row   = tile * 16 + (lane & 15);
    int klane = (lane >> 4) << 1;

    v8f acc[4];
#pragma unroll
    for (int t = 0; t < 4; ++t) {
        float bv = brel[t * 16 + (lane & 15)];
        acc[t] = (v8f){bv, bv, bv, bv, bv, bv, bv, bv};
    }

    const float* arow = agg + (size_t)row * D;
    const float* hrow = h   + (size_t)row * D;

    for (int k0 = 0; k0 < D; k0 += 4) {
        int kk = k0 + klane;
        v2f aa = *(const v2f*)(arow + kk);
        v2f ah = *(const v2f*)(hrow + kk);
#pragma unroll
        for (int t = 0; t < 4; ++t) {
            int col = t * 16 + (lane & 15);
            v2f b0 = *(const v2f*)&lw[0 * D * LDP + col * LDP + kk];
            v2f b1 = *(const v2f*)&lw[1 * D * LDP + col * LDP + kk];
            acc[t] = __builtin_amdgcn_wmma_f32_16x16x4_f32(false, aa, false, b0, (short)0, acc[t], false, false);
            acc[t] = __builtin_amdgcn_wmma_f32_16x16x4_f32(false, ah, false, b1, (short)0, acc[t], false, false);
        }
    }

    int mo = (lane >> 4) * 8;
#pragma unroll
    for (int t = 0; t < 4; ++t) {
        int col = t * 16 + (lane & 15);
#pragma unroll
        for (int v = 0; v < 8; ++v)
            out[(size_t)(tile * 16 + mo + v) * D + col] = acc[t][v];
    }
}

extern "C" void kernel_launch(void* const* d_in, const int* in_sizes, int n_in,
                              void* d_out, int out_size, void* d_ws, size_t ws_size,
                              hipStream_t stream) {
    const float* x       = (const float*)d_in[0];
    const int*   ei      = (const int*)d_in[1];   // [2][E]
    const int*   et      = (const int*)d_in[2];   // [E]
    const float* rgcn_w  = (const float*)d_in[3]; // [2][64][64]
    const float* rgcn_r  = (const float*)d_in[4]; // [64][64]
    const float* rgcn_b  = (const float*)d_in[5]; // [64]
    const float* gc_wrel = (const float*)d_in[6]; // [64][64]
    const float* gc_brel = (const float*)d_in[7]; // [64]
    const float* gc_wroot= (const float*)d_in[8]; // [64][64]
    float* out = (float*)d_out;                   // doubles as h between stages

    float* sm  = (float*)d_ws;                        // 2*N*64
    float* cnt = sm + (size_t)2 * N_NODES * D;        // 2*N
    float* agg = cnt + (size_t)2 * N_NODES;           // N*64
    size_t zf = (size_t)2 * N_NODES * D + (size_t)2 * N_NODES + (size_t)N_NODES * D;
    int n4 = (int)(zf / 4);

    rgcn_zero<<<(n4 + 255) / 256, 256, 0, stream>>>((float*)d_ws, n4);
    rgcn_scatter<<<N_EDGES / 8, 256, 0, stream>>>(x, ei, et, sm, cnt);
    rgcn_gemm<<<(NT + 7) / 8, 256, 0, stream>>>(x, sm, cnt, rgcn_w, rgcn_r, rgcn_b, out);
    gc_scatter<<<N_EDGES / 8, 256, 0, stream>>>(out, ei, agg);
    gc_gemm<<<(NT + 7) / 8, 256, 0, stream>>>(agg, out, gc_wrel, gc_wroot, gc_brel, out);
}